// _KoopmanLayer_88759794139276
// MI455X (gfx1250) — compile-verified
//
#include <hip/hip_runtime.h>
#include <hip/hip_bf16.h>

#define DIM     64
#define TSTEPS  1000
#define BATCHSZ 1024
#define ROWSTRIDE (TSTEPS * DIM)   // elements between consecutive b rows: 64000

typedef __attribute__((ext_vector_type(2))) float v2f;
typedef __attribute__((ext_vector_type(8))) float v8f;

// ---------------------------------------------------------------------------
// 64x64 f32 matmul helper: C = A * B (row-major), 256 threads, caller syncs.
// Each thread computes one row-slice of 16 outputs.
// ---------------------------------------------------------------------------
__device__ inline void mm64(const float* __restrict__ A,
                            const float* __restrict__ B,
                            float* __restrict__ C, int tid) {
    const int row = tid >> 2;          // 0..63
    const int c0  = (tid & 3) * 16;    // 0,16,32,48
    float acc[16];
#pragma unroll
    for (int c = 0; c < 16; ++c) acc[c] = 0.f;
    for (int j = 0; j < DIM; ++j) {
        const float a = A[row * DIM + j];
        const float* Brow = B + j * DIM + c0;
#pragma unroll
        for (int c = 0; c < 16; ++c) acc[c] = fmaf(a, Brow[c], acc[c]);
    }
#pragma unroll
    for (int c = 0; c < 16; ++c) C[row * DIM + c0 + c] = acc[c];
}

// ---------------------------------------------------------------------------
// Kernel 1: M = expm(K) via Taylor series (||K|| ~ 0.25 -> 24 terms is exact
// to fp32 epsilon).  One workgroup.
// ---------------------------------------------------------------------------
__global__ void __launch_bounds__(256) expm_kernel(const float* __restrict__ Kmat,
                                                   float* __restrict__ M) {
    __shared__ float sK[DIM * DIM];
    __shared__ float sT[DIM * DIM];
    __shared__ float sA[DIM * DIM];
    __shared__ float sTmp[DIM * DIM];
    const int tid = threadIdx.x;
    for (int e = tid; e < DIM * DIM; e += 256) {
        const float k = Kmat[e];
        const int r = e / DIM, c = e % DIM;
        sK[e] = k;
        sT[e] = k;                       // term_1 = K
        sA[e] = ((r == c) ? 1.f : 0.f) + k;  // I + K
    }
    __syncthreads();
    for (int n = 2; n <= 24; ++n) {
        mm64(sT, sK, sTmp, tid);         // tmp = term_{n-1} * K
        __syncthreads();
        const float inv = 1.f / (float)n;
        for (int e = tid; e < DIM * DIM; e += 256) {
            const float v = sTmp[e] * inv;   // term_n = tmp / n
            sT[e] = v;
            sA[e] += v;
        }
        __syncthreads();
    }
    for (int e = tid; e < DIM * DIM; e += 256) M[e] = sA[e];
}

// ---------------------------------------------------------------------------
// Kernel 2: R[r] = M^r for r=0..32, Qp[k] = (M^32)^k for k=0..31.
// One workgroup, sequential chain of 63 tiny matmuls.
// ---------------------------------------------------------------------------
__global__ void __launch_bounds__(256) powers_kernel(const float* __restrict__ M,
                                                     float* __restrict__ R,
                                                     float* __restrict__ Qp) {
    __shared__ float sM[DIM * DIM];
    __shared__ float sP[DIM * DIM];
    __shared__ float sTmp[DIM * DIM];
    const int tid = threadIdx.x;
    for (int e = tid; e < DIM * DIM; e += 256) {
        const int r = e / DIM, c = e % DIM;
        sM[e] = M[e];
        sP[e] = (r == c) ? 1.f : 0.f;
        R[e]  = sP[e];                       // R[0] = I
    }
    __syncthreads();
    for (int p = 1; p <= 32; ++p) {
        mm64(sP, sM, sTmp, tid);
        __syncthreads();
        for (int e = tid; e < DIM * DIM; e += 256) {
            sP[e] = sTmp[e];
            R[p * DIM * DIM + e] = sTmp[e];
        }
        __syncthreads();
    }
    // sP now holds M^32; restart chain for Qp
    for (int e = tid; e < DIM * DIM; e += 256) {
        const int r = e / DIM, c = e % DIM;
        sM[e] = sP[e];                       // sM = M^32
        const float id = (r == c) ? 1.f : 0.f;
        sP[e] = id;
        Qp[e] = id;                          // Qp[0] = I
    }
    __syncthreads();
    for (int p = 1; p < 32; ++p) {
        mm64(sP, sM, sTmp, tid);
        __syncthreads();
        for (int e = tid; e < DIM * DIM; e += 256) {
            sP[e] = sTmp[e];
            Qp[p * DIM * DIM + e] = sTmp[e];
        }
        __syncthreads();
    }
}

// ---------------------------------------------------------------------------
// Kernel 3: PT[t] = transpose(Qp[t/32] * R[t%32]).  1000 independent WGs.
// Transposed storage makes WMMA B-fragment loads lane-contiguous.
// ---------------------------------------------------------------------------
__global__ void __launch_bounds__(256) buildPT_kernel(const float* __restrict__ R,
                                                      const float* __restrict__ Qp,
                                                      float* __restrict__ PT) {
    const int t = blockIdx.x;
    const int k = t >> 5;      // t / 32
    const int r = t & 31;      // t % 32
    __shared__ float sQ[DIM * DIM];
    __shared__ float sR[DIM * DIM];
    __shared__ float sP[DIM * DIM];
    const int tid = threadIdx.x;
    for (int e = tid; e < DIM * DIM; e += 256) {
        sQ[e] = Qp[k * DIM * DIM + e];
        sR[e] = R[r * DIM * DIM + e];
    }
    __syncthreads();
    mm64(sQ, sR, sP, tid);     // P_t = Q^k * M^r  (commuting powers of M)
    __syncthreads();
    float* dst = PT + (size_t)t * DIM * DIM;
    for (int e = tid; e < DIM * DIM; e += 256) {
        const int i = e / DIM, j = e % DIM;
        dst[j * DIM + i] = sP[e];          // store transposed: PT[t][j][i]
    }
}

// ---------------------------------------------------------------------------
// Kernel 4: out[b,t,i] = sum_j PT[t][j][i] * x0[b][j]  via f32 WMMA 16x16x4.
// Mapping: WMMA M -> batch, N -> i, K -> j.
//   A (16x4 f32): lane l holds row b = b_tile + (l&15), K = 2*(l>>4)+vgpr
//   B (4x16 f32): lane l holds col i = i_tile + (l&15), K = 2*(l>>4)+vgpr
//   C/D (16x16):  VGPR r, lane l -> b = r + 8*(l>>4), i = (l&15)
// Each wave: 32 b x 64 i for one t; 8 waves/WG; grid = (1024/256) x 1000.
// ---------------------------------------------------------------------------
__global__ void __launch_bounds__(256) koopman_apply(const float* __restrict__ x0,
                                                     const float* __restrict__ PT,
                                                     float* __restrict__ out) {
    const int t    = blockIdx.y;
    const int lane = threadIdx.x & 31;
    const int wave = threadIdx.x >> 5;
    const int half = lane >> 4;    // 0 or 1
    const int lm   = lane & 15;
    const int b0   = blockIdx.x * 256 + wave * 32;

    // Preload A fragments: two 16-row batch tiles, all 16 K-steps.
    v2f a[2][16];
#pragma unroll
    for (int bt = 0; bt < 2; ++bt) {
        const float* xr = x0 + (size_t)(b0 + bt * 16 + lm) * ROWSTRIDE + 2 * half;
#pragma unroll
        for (int k = 0; k < 16; ++k)
            a[bt][k] = *(const v2f*)(xr + 4 * k);
    }

    v8f acc[2][4];
#pragma unroll
    for (int bt = 0; bt < 2; ++bt)
#pragma unroll
        for (int n = 0; n < 4; ++n)
            acc[bt][n] = (v8f){0.f, 0.f, 0.f, 0.f, 0.f, 0.f, 0.f, 0.f};

    // B fragments stream from L2-resident transposed powers.
    const float* Pb = PT + (size_t)t * DIM * DIM + (2 * half) * DIM + lm;
#pragma unroll
    for (int k = 0; k < 16; ++k) {
        const float* pj = Pb + 4 * k * DIM;    // j = 4k + 2*half (+vgpr)
#pragma unroll
        for (int n = 0; n < 4; ++n) {
            v2f bf;
            bf.x = pj[n * 16];                 // K = 2*half,   i = 16n + lm
            bf.y = pj[DIM + n * 16];           // K = 2*half+1
            acc[0][n] = __builtin_amdgcn_wmma_f32_16x16x4_f32(
                false, a[0][k], false, bf, (short)0, acc[0][n], false, false);
            acc[1][n] = __builtin_amdgcn_wmma_f32_16x16x4_f32(
                false, a[1][k], false, bf, (short)0, acc[1][n], false, false);
        }
    }

    // Store: per C VGPR, the 16-lane halves write contiguous 64B segments.
#pragma unroll
    for (int bt = 0; bt < 2; ++bt) {
        float* ob = out + (size_t)(b0 + bt * 16 + 8 * half) * ROWSTRIDE
                        + (size_t)t * DIM + lm;
#pragma unroll
        for (int n = 0; n < 4; ++n)
#pragma unroll
            for (int r = 0; r < 8; ++r)
                ob[(size_t)r * ROWSTRIDE + n * 16] = acc[bt][n][r];
    }
}

// ---------------------------------------------------------------------------
// Host launcher.  d_in[0] = inputs (1024,1000,64) f32, d_in[1] = kernel (64,64).
// Workspace layout (floats): M[4096] | R[33*4096] | Qp[32*4096] | PT[1000*4096]
// total ~17.5 MB.
// ---------------------------------------------------------------------------
extern "C" void kernel_launch(void* const* d_in, const int* in_sizes, int n_in,
                              void* d_out, int out_size, void* d_ws, size_t ws_size,
                              hipStream_t stream) {
    (void)in_sizes; (void)n_in; (void)out_size; (void)ws_size;
    const float* inp = (const float*)d_in[0];
    const float* Km  = (const float*)d_in[1];
    float* out = (float*)d_out;
    float* ws  = (float*)d_ws;

    float* M  = ws;                       //  1 * 4096
    float* R  = ws + 4096;                // 33 * 4096
    float* Qp = ws + 4096 * 34;           // 32 * 4096
    float* PT = ws + 4096 * 66;           // 1000 * 4096

    expm_kernel<<<1, 256, 0, stream>>>(Km, M);
    powers_kernel<<<1, 256, 0, stream>>>(M, R, Qp);
    buildPT_kernel<<<TSTEPS, 256, 0, stream>>>(R, Qp, PT);

    dim3 grid(BATCHSZ / 256, TSTEPS);
    koopman_apply<<<grid, 256, 0, stream>>>(inp, PT, out);
}